// Moment_Probing_ViT_59992103190766
// MI455X (gfx1250) — compile-verified
//
#include <hip/hip_runtime.h>
#include <hip/hip_bf16.h>
#include <math.h>
#include <stdint.h>

typedef float v2f __attribute__((ext_vector_type(2)));
typedef float v8f __attribute__((ext_vector_type(8)));
typedef int   v4i __attribute__((ext_vector_type(4)));

#define B_   256
#define LP   196
#define D_   768
#define HD   512
#define DH   128
#define NC   1000
#define SA_STRIDE 772   // 768 + 4 pad: stride%64banks==4 -> conflict-free A fragment reads

// ---- CDNA5 async global->LDS path (guarded; falls back to register copy) ----
#if defined(__AMDGCN__) && __has_builtin(__builtin_amdgcn_global_load_async_to_lds_b128)
#define ASYNC_LDS 1
typedef __attribute__((address_space(1))) v4i gv4i_t;   // global int4*
typedef __attribute__((address_space(3))) v4i lv4i_t;   // LDS int4*
__device__ __forceinline__ void async_wait0() {
#if __has_builtin(__builtin_amdgcn_s_wait_asynccnt)
  __builtin_amdgcn_s_wait_asynccnt(0);
#else
  asm volatile("s_wait_asynccnt 0" ::: "memory");
#endif
}
#endif

// ---------------- utility ----------------
__global__ void k_zero(float* p, int n) {
  int i = blockIdx.x * blockDim.x + threadIdx.x;
  if (i < n) p[i] = 0.f;
}

// ---------------- stage 1: projection GEMM (WMMA f32 16x16x4) ----------------
// X: logical (50176 x 768), row r = b*196+l reads x[b, l+1, :]
// A-tile (16x768) staged once per block in LDS via async loads; 8 waves each
// compute a 16x64 tile of the 16x512 block output.
__global__ __launch_bounds__(256) void k_proj(const float* __restrict__ x,
                                              const float* __restrict__ w,
                                              const float* __restrict__ bias,
                                              float* __restrict__ xp) {
  __shared__ float sA[16 * SA_STRIDE];
  const int row0 = blockIdx.x * 16;

  { // ---- stage A tile: 16 threads per row, 12 x b128 chunks per thread ----
    const int rowA = threadIdx.x >> 4;   // 0..15
    const int ch   = threadIdx.x & 15;   // 0..15
    const int row  = row0 + rowA;
    const int bidx = row / LP;
    const int l    = row - bidx * LP;
    const float* src = x + ((size_t)bidx * 197 + l + 1) * D_;
    float* dst = sA + rowA * SA_STRIDE;
#ifdef ASYNC_LDS
#pragma unroll
    for (int it = 0; it < 12; ++it) {
      int c4 = (ch + it * 16) * 4;
      __builtin_amdgcn_global_load_async_to_lds_b128(
          (gv4i_t*)(uintptr_t)(src + c4),
          (lv4i_t*)(uintptr_t)(dst + c4), 0, 0);
    }
    async_wait0();
#else
#pragma unroll
    for (int it = 0; it < 12; ++it) {
      int c4 = (ch + it * 16) * 4;
      *(float4*)(dst + c4) = *(const float4*)(src + c4);
    }
#endif
    __syncthreads();
  }

  const int t    = threadIdx.x & 31;
  const int wave = threadIdx.x >> 5;
  const int col0 = wave * 64;
  const int m  = t & 15;            // A: M index
  const int kb = (t >> 4) * 2;      // A/B: K pair base (0 or 2)
  const int n  = t & 15;            // B/C/D: N index
  const float* aRow = sA + m * SA_STRIDE;

  v8f acc0 = {}, acc1 = {}, acc2 = {}, acc3 = {};
  for (int k0 = 0; k0 < D_; k0 += 4) {
    const float* wk = w + (size_t)(k0 + kb) * HD;
    if ((k0 & 63) == 0) __builtin_prefetch(wk + 16 * HD, 0, 0);
    v2f a = *(const v2f*)(aRow + k0 + kb);           // ds_load_b64, conflict-free
    v2f b0, b1, b2, b3;
    b0.x = wk[col0      + n]; b0.y = wk[HD + col0      + n];
    b1.x = wk[col0 + 16 + n]; b1.y = wk[HD + col0 + 16 + n];
    b2.x = wk[col0 + 32 + n]; b2.y = wk[HD + col0 + 32 + n];
    b3.x = wk[col0 + 48 + n]; b3.y = wk[HD + col0 + 48 + n];
    acc0 = __builtin_amdgcn_wmma_f32_16x16x4_f32(false, a, false, b0, (short)0, acc0, false, false);
    acc1 = __builtin_amdgcn_wmma_f32_16x16x4_f32(false, a, false, b1, (short)0, acc1, false, false);
    acc2 = __builtin_amdgcn_wmma_f32_16x16x4_f32(false, a, false, b2, (short)0, acc2, false, false);
    acc3 = __builtin_amdgcn_wmma_f32_16x16x4_f32(false, a, false, b3, (short)0, acc3, false, false);
  }
  // C/D layout: VGPR j -> M = j + 8*(lane>=16), N = lane&15
  const int mb = (t >> 4) * 8;
  float bi0 = bias[col0 + n], bi1 = bias[col0 + 16 + n];
  float bi2 = bias[col0 + 32 + n], bi3 = bias[col0 + 48 + n];
#pragma unroll
  for (int j = 0; j < 8; ++j) {
    float* orow = xp + (size_t)(row0 + mb + j) * HD + col0;
    orow[n]      = acc0[j] + bi0;
    orow[16 + n] = acc1[j] + bi1;
    orow[32 + n] = acc2[j] + bi2;
    orow[48 + n] = acc3[j] + bi3;
  }
}

// ---------------- stage 2: per-row LayerNorm + mean accumulation ----------------
__global__ __launch_bounds__(256) void k_ln(float* __restrict__ buf,
                                            const float* __restrict__ g,
                                            const float* __restrict__ bb,
                                            float* __restrict__ meanbuf) {
  const int wave = threadIdx.x >> 5;
  const int lane = threadIdx.x & 31;
  const int R = blockIdx.x * 8 + wave;        // < 200704
  const int r = R >> 2;
  const int h = R & 3;
  float* row = buf + (size_t)r * HD + h * DH;
  float4 v = *(float4*)(row + lane * 4);
  float s  = v.x + v.y + v.z + v.w;
  float sq = v.x * v.x + v.y * v.y + v.z * v.z + v.w * v.w;
#pragma unroll
  for (int off = 16; off > 0; off >>= 1) {
    s  += __shfl_xor(s,  off, 32);
    sq += __shfl_xor(sq, off, 32);
  }
  const float mu  = s * (1.f / DH);
  const float var = sq * (1.f / DH) - mu * mu;
  const float inv = rsqrtf(var + 1e-5f);
  float4 gg = *(const float4*)(g  + lane * 4);
  float4 bv = *(const float4*)(bb + lane * 4);
  float4 y;
  y.x = (v.x - mu) * inv * gg.x + bv.x;
  y.y = (v.y - mu) * inv * gg.y + bv.y;
  y.z = (v.z - mu) * inv * gg.z + bv.z;
  y.w = (v.w - mu) * inv * gg.w + bv.w;
  *(float4*)(row + lane * 4) = y;
  float rs = y.x + y.y + y.z + y.w;
#pragma unroll
  for (int off = 16; off > 0; off >>= 1) rs += __shfl_xor(rs, off, 32);
  if (lane == 0) {
    int bidx = r / LP;
    atomicAdd(meanbuf + h * B_ + bidx, rs);
  }
}

// ---------------- stage 3: subtract per-(h,b) global mean ----------------
__global__ __launch_bounds__(256) void k_sub(float* __restrict__ buf,
                                             const float* __restrict__ meanbuf) {
  size_t idx = (size_t)blockIdx.x * 256 + threadIdx.x;   // < 25690112
  int r   = (int)(idx >> 9);
  int col = (int)(idx & 511);
  int h   = col >> 7;
  int b   = r / LP;
  buf[idx] -= meanbuf[h * B_ + b] * (1.f / (LP * DH));
}

// ---------------- stage 4: covariance GEMMs (WMMA f32 16x16x4) ----------------
__global__ __launch_bounds__(256) void k_cov(const float* __restrict__ buf,
                                             float* __restrict__ cov) {
  const int t    = threadIdx.x & 31;
  const int wave = threadIdx.x >> 5;
  const int d0 = blockIdx.x * 16;
  const int b  = blockIdx.y;
  const int p  = blockIdx.z;
  const int e0 = wave * 16;
  const int m  = t & 15;
  const int kb = (t >> 4) * 2;
  const float* b1 = buf + (size_t)b * LP * HD + p * DH;
  const float* b2 = b1 + DH;
  v8f acc = {};
  for (int k0 = 0; k0 < LP; k0 += 4) {
    const float* r0 = b1 + (size_t)(k0 + kb) * HD;
    const float* r1 = b2 + (size_t)(k0 + kb) * HD;
    v2f a, bv;
    a.x  = r0[d0 + m];  a.y  = r0[HD + d0 + m];
    bv.x = r1[e0 + m];  bv.y = r1[HD + e0 + m];
    acc = __builtin_amdgcn_wmma_f32_16x16x4_f32(false, a, false, bv, (short)0, acc, false, false);
  }
  const int mb = (t >> 4) * 8;
  const int n  = t & 15;
  float* co = cov + ((size_t)p * B_ + b) * DH * DH;
#pragma unroll
  for (int j = 0; j < 8; ++j)
    co[(size_t)(d0 + mb + j) * DH + e0 + n] = acc[j] * (1.f / LP);
}

// ---------------- stage 5: column L2-normalization ----------------
__global__ __launch_bounds__(128) void k_colnorm(float* __restrict__ cov) {
  const int pb = blockIdx.x;          // p*256 + b
  const int e  = threadIdx.x;
  float* c = cov + (size_t)pb * DH * DH;
  float s = 0.f;
  for (int d = 0; d < DH; ++d) { float v = c[d * DH + e]; s += v * v; }
  const float scale = 1.f / fmaxf(sqrtf(s), 1e-12f);
  for (int d = 0; d < DH; ++d) c[d * DH + e] *= scale;
}

// ---------------- stage 6: conv1 (3x3 stride2 VALID) + exact GELU ----------------
__global__ __launch_bounds__(256) void k_conv1(const float* __restrict__ cov,
                                               const float* __restrict__ w1,
                                               float* __restrict__ z1) {
  int idx = blockIdx.x * 256 + threadIdx.x;
  if (idx >= 256 * 3 * 63 * 63) return;
  int j = idx % 63;
  int i = (idx / 63) % 63;
  int o = (idx / (63 * 63)) % 3;
  int b = idx / (63 * 63 * 3);
  float acc = 0.f;
#pragma unroll
  for (int c = 0; c < 3; ++c) {
    const float* in = cov + ((size_t)c * B_ + b) * DH * DH;   // cov[c][b]
#pragma unroll
    for (int kh = 0; kh < 3; ++kh)
#pragma unroll
      for (int kw = 0; kw < 3; ++kw)
        acc += w1[((o * 3 + c) * 3 + kh) * 3 + kw] * in[(2 * i + kh) * DH + 2 * j + kw];
  }
  float gl = 0.5f * acc * (1.f + erff(acc * 0.70710678118654752f));
  z1[((size_t)(b * 3 + o) * 63 + i) * 63 + j] = gl;
}

// ---------------- stage 7: conv2 (3x3 stride2 VALID) ----------------
__global__ __launch_bounds__(256) void k_conv2(const float* __restrict__ z1,
                                               const float* __restrict__ w2,
                                               float* __restrict__ z2) {
  int idx = blockIdx.x * 256 + threadIdx.x;
  if (idx >= 256 * 3 * 31 * 31) return;
  int j = idx % 31;
  int i = (idx / 31) % 31;
  int o = (idx / (31 * 31)) % 3;
  int b = idx / (31 * 31 * 3);
  float acc = 0.f;
#pragma unroll
  for (int c = 0; c < 3; ++c) {
    const float* in = z1 + (size_t)(b * 3 + c) * 63 * 63;
#pragma unroll
    for (int kh = 0; kh < 3; ++kh)
#pragma unroll
      for (int kw = 0; kw < 3; ++kw)
        acc += w2[((o * 3 + c) * 3 + kh) * 3 + kw] * in[(2 * i + kh) * 63 + 2 * j + kw];
  }
  z2[((size_t)(b * 3 + o) * 31 + i) * 31 + j] = acc;    // == flat[b][o*961 + i*31 + j]
}

// ---------------- stage 8: fused classifier head (WMMA f32) ----------------
// out[b,c] = 0.5*( cls[b,:]@W1[:,c] + flat[b,:]@W2[:,c] + b1[c] + b2[c] )
// grid (2, 16): x -> 512-col slab, y -> 16-row tile. wave -> 16x64 tile.
// K2 = 2883 = 720*4 + 3; odd tail folded into the epilogue as scalar FMAs.
__global__ __launch_bounds__(256) void k_head(const float* __restrict__ cls,
                                              const float* __restrict__ w1,
                                              const float* __restrict__ b1,
                                              const float* __restrict__ flat,
                                              const float* __restrict__ w2,
                                              const float* __restrict__ b2,
                                              float* __restrict__ out) {
  const int t    = threadIdx.x & 31;
  const int wave = threadIdx.x >> 5;
  const int row0    = blockIdx.y * 16;
  const int colBase = blockIdx.x * 512 + wave * 64;
  const int m  = t & 15;
  const int kb = (t >> 4) * 2;
  const int n  = t & 15;
  const int row = row0 + m;
  // clamped load columns (stores are guarded, so clamped dup loads are harmless)
  const int s0 = min(colBase      + n, NC - 1);
  const int s1 = min(colBase + 16 + n, NC - 1);
  const int s2 = min(colBase + 32 + n, NC - 1);
  const int s3 = min(colBase + 48 + n, NC - 1);

  v8f acc0 = {}, acc1 = {}, acc2 = {}, acc3 = {};

  const float* arow = cls + (size_t)row * D_;
  for (int k0 = 0; k0 < D_; k0 += 4) {
    v2f a = *(const v2f*)(arow + k0 + kb);
    const float* wk = w1 + (size_t)(k0 + kb) * NC;
    v2f b0v, b1v, b2v, b3v;
    b0v.x = wk[s0]; b0v.y = wk[NC + s0];
    b1v.x = wk[s1]; b1v.y = wk[NC + s1];
    b2v.x = wk[s2]; b2v.y = wk[NC + s2];
    b3v.x = wk[s3]; b3v.y = wk[NC + s3];
    acc0 = __builtin_amdgcn_wmma_f32_16x16x4_f32(false, a, false, b0v, (short)0, acc0, false, false);
    acc1 = __builtin_amdgcn_wmma_f32_16x16x4_f32(false, a, false, b1v, (short)0, acc1, false, false);
    acc2 = __builtin_amdgcn_wmma_f32_16x16x4_f32(false, a, false, b2v, (short)0, acc2, false, false);
    acc3 = __builtin_amdgcn_wmma_f32_16x16x4_f32(false, a, false, b3v, (short)0, acc3, false, false);
  }

  const float* frow = flat + (size_t)row * 2883;
  for (int k0 = 0; k0 < 2880; k0 += 4) {
    v2f a;  // row stride 2883 is odd -> scalar loads (no 8B-aligned pair)
    a.x = frow[k0 + kb]; a.y = frow[k0 + kb + 1];
    const float* wk = w2 + (size_t)(k0 + kb) * NC;
    if ((k0 & 63) == 0) __builtin_prefetch(wk + 16 * NC, 0, 0);
    v2f b0v, b1v, b2v, b3v;
    b0v.x = wk[s0]; b0v.y = wk[NC + s0];
    b1v.x = wk[s1]; b1v.y = wk[NC + s1];
    b2v.x = wk[s2]; b2v.y = wk[NC + s2];
    b3v.x = wk[s3]; b3v.y = wk[NC + s3];
    acc0 = __builtin_amdgcn_wmma_f32_16x16x4_f32(false, a, false, b0v, (short)0, acc0, false, false);
    acc1 = __builtin_amdgcn_wmma_f32_16x16x4_f32(false, a, false, b1v, (short)0, acc1, false, false);
    acc2 = __builtin_amdgcn_wmma_f32_16x16x4_f32(false, a, false, b2v, (short)0, acc2, false, false);
    acc3 = __builtin_amdgcn_wmma_f32_16x16x4_f32(false, a, false, b3v, (short)0, acc3, false, false);
  }

  const int mb = (t >> 4) * 8;
#define HEAD_STORE(ACC, G)                                                     \
  do {                                                                         \
    int c = colBase + (G) * 16 + n;                                            \
    if (c < NC) {                                                              \
      float wt0 = w2[(size_t)2880 * NC + c];                                   \
      float wt1 = w2[(size_t)2881 * NC + c];                                   \
      float wt2 = w2[(size_t)2882 * NC + c];                                   \
      float bsum = b1[c] + b2[c];                                              \
      _Pragma("unroll") for (int j = 0; j < 8; ++j) {                          \
        int r = row0 + mb + j;                                                 \
        const float* fr = flat + (size_t)r * 2883;                             \
        float v = ACC[j] + fr[2880] * wt0 + fr[2881] * wt1 + fr[2882] * wt2;   \
        out[(size_t)r * NC + c] = 0.5f * (v + bsum);                           \
      }                                                                        \
    }                                                                          \
  } while (0)
  HEAD_STORE(acc0, 0);
  HEAD_STORE(acc1, 1);
  HEAD_STORE(acc2, 2);
  HEAD_STORE(acc3, 3);
#undef HEAD_STORE
}

// ---------------- launcher ----------------
extern "C" void kernel_launch(void* const* d_in, const int* in_sizes, int n_in,
                              void* d_out, int out_size, void* d_ws, size_t ws_size,
                              hipStream_t stream) {
  const float* cls_token = (const float*)d_in[0];
  const float* x        = (const float*)d_in[1];
  const float* proj_w   = (const float*)d_in[2];
  const float* proj_b   = (const float*)d_in[3];
  const float* ln_g     = (const float*)d_in[4];
  const float* ln_b     = (const float*)d_in[5];
  const float* conv1_w  = (const float*)d_in[6];
  const float* conv2_w  = (const float*)d_in[7];
  const float* cls1_w   = (const float*)d_in[8];
  const float* cls1_b   = (const float*)d_in[9];
  const float* cls2_w   = (const float*)d_in[10];
  const float* cls2_b   = (const float*)d_in[11];
  float* out = (float*)d_out;

  float* ws = (float*)d_ws;
  const size_t N_XP   = (size_t)50176 * HD;            // 25,690,112
  const size_t N_MEAN = 4 * B_;                        // 1,024
  const size_t N_COV  = (size_t)3 * B_ * DH * DH;      // 12,582,912
  const size_t N_Z1   = (size_t)B_ * 3 * 63 * 63;      // 3,048,192
  float* xp   = ws;
  float* mean = xp + N_XP;
  float* cov  = mean + N_MEAN;
  float* z1   = cov + N_COV;
  float* z2   = z1 + N_Z1;                             // 256*2883 floats

  k_zero<<<4, 256, 0, stream>>>(mean, (int)N_MEAN);
  k_proj<<<3136, 256, 0, stream>>>(x, proj_w, proj_b, xp);
  k_ln<<<25088, 256, 0, stream>>>(xp, ln_g, ln_b, mean);
  k_sub<<<100352, 256, 0, stream>>>(xp, mean);
  k_cov<<<dim3(8, 256, 3), 256, 0, stream>>>(xp, cov);
  k_colnorm<<<768, 128, 0, stream>>>(cov);
  k_conv1<<<(256 * 3 * 63 * 63 + 255) / 256, 256, 0, stream>>>(cov, conv1_w, z1);
  k_conv2<<<(256 * 3 * 31 * 31 + 255) / 256, 256, 0, stream>>>(z1, conv2_w, z2);
  k_head<<<dim3(2, 16), 256, 0, stream>>>(cls_token, cls1_w, cls1_b, z2, cls2_w, cls2_b, out);
}